// laa_model_19679540150905
// MI455X (gfx1250) — compile-verified
//
#include <hip/hip_runtime.h>

#define B_DIM   16384
#define SRC_DIM 200
#define C_DIM   10
#define I_DIM   2000          // SRC*C
#define EPSF    1e-10f

typedef __attribute__((ext_vector_type(2))) float v2f;
typedef __attribute__((ext_vector_type(8))) float v8f;

// ---------------------------------------------------------------------------
// Kernel 1: build WMMA B-operands in row-pair layout.
//   BrecP[(k/2)*16 + n] = ( -logp[n][k]/I , -logp[n][k+1]/I )   (n<10, else 0)
//   BclsP[(k/2)*16 + n] = (  W_yc[k][n]   ,  W_yc[k+1][n]   )   (n<10, else 0)
// k and k+1 always share a source block (C=10 even), so one softmax sum each.
// ---------------------------------------------------------------------------
__global__ void prep_b_kernel(const float* __restrict__ W_reconstr,  // [C,I]
                              const float* __restrict__ b_reconstr,  // [I]
                              const float* __restrict__ W_yc,        // [I,C]
                              float2* __restrict__ BrecP,            // [I/2*16]
                              float2* __restrict__ BclsP) {
    int t = blockIdx.x * blockDim.x + threadIdx.x;
    const int total = (I_DIM / 2) * 16;                // 16000
    if (t >= total) return;
    int col = t & 15;
    int k   = (t >> 4) * 2;
    float2 rec = make_float2(0.f, 0.f);
    float2 cls = make_float2(0.f, 0.f);
    if (col < C_DIM) {
        cls.x = W_yc[(k + 0) * C_DIM + col];
        cls.y = W_yc[(k + 1) * C_DIM + col];
        int base = (k / C_DIM) * C_DIM;                // source-block start
        float lg[C_DIM];
        float S = 0.f;
        #pragma unroll
        for (int j = 0; j < C_DIM; ++j) {
            float L = W_reconstr[col * I_DIM + base + j] + b_reconstr[base + j];
            lg[j] = L;
            S += __expf(L);
        }
        int j0 = k - base;
        float p0 = __expf(lg[j0])     / S;
        float p1 = __expf(lg[j0 + 1]) / S;
        rec.x = -__logf(EPSF + p0) * (1.0f / I_DIM);
        rec.y = -__logf(EPSF + p1) * (1.0f / I_DIM);
    }
    BrecP[t] = rec;
    BclsP[t] = cls;
}

// ---------------------------------------------------------------------------
// Kernel 2: fused dual-GEMM over x / x*mask with V_WMMA_F32_16X16X4_F32,
// plus softmax / per-row loss epilogue. One wave = 16 batch rows.
// ---------------------------------------------------------------------------
__launch_bounds__(256)
__global__ void fused_main_kernel(const float*  __restrict__ x,
                                  const float*  __restrict__ mask,
                                  const float*  __restrict__ y_target,  // [B,C]
                                  const float*  __restrict__ b_yc,      // [C]
                                  const float2* __restrict__ BrecP,
                                  const float2* __restrict__ BclsP,
                                  float*        __restrict__ y_out,     // [B,C]
                                  float*        __restrict__ block_partials) {
    const int lane = threadIdx.x & 31;
    const int wave = threadIdx.x >> 5;
    const int col  = lane & 15;         // A-matrix: M = lane&15 for all lanes
    const int half = lane >> 4;         // K offset: lanes 16-31 hold K+2,K+3
    const int rowBase = (blockIdx.x * 8 + wave) * 16;

    const float*  xrow = x    + (size_t)(rowBase + col) * I_DIM + 2 * half;
    const float*  mrow = mask + (size_t)(rowBase + col) * I_DIM + 2 * half;
    const float2* br   = BrecP + lane;  // pair layout: index = lane + 8*k0
    const float2* bc   = BclsP + lane;

    v8f acc_rec = {};
    v8f acc_cls = {};

    for (int kb = 0; kb < I_DIM; kb += 40) {
        __builtin_prefetch(xrow + kb + 320, 0, 3);   // global_prefetch_b8
        __builtin_prefetch(mrow + kb + 320, 0, 3);
        #pragma unroll
        for (int u = 0; u < 10; ++u) {
            const int k0 = kb + 4 * u;
            float2 xv = *(const float2*)(xrow + k0);
            float2 mv = *(const float2*)(mrow + k0);
            float2 bv = br[8 * k0];
            float2 cv = bc[8 * k0];
            v2f a_cls = {xv.x,           xv.y};
            v2f a_rec = {xv.x * mv.x,    xv.y * mv.y};
            v2f b_rec = {bv.x,           bv.y};
            v2f b_cls = {cv.x,           cv.y};
            acc_rec = __builtin_amdgcn_wmma_f32_16x16x4_f32(
                false, a_rec, false, b_rec, (short)0, acc_rec, false, false);
            acc_cls = __builtin_amdgcn_wmma_f32_16x16x4_f32(
                false, a_cls, false, b_cls, (short)0, acc_cls, false, false);
        }
    }

    // ---- spill D tiles to per-wave LDS (D layout: VGPR r -> row r+8*half) --
    __shared__ float sm[8][2][16][17];
    __shared__ float wavesum[8];
    #pragma unroll
    for (int r = 0; r < 8; ++r) {
        sm[wave][0][r + 8 * half][col] = acc_rec[r];
        sm[wave][1][r + 8 * half][col] = acc_cls[r];
    }
    asm volatile("s_wait_dscnt 0" ::: "memory");

    float contrib = 0.f;
    if (lane < 16) {
        const int gr = rowBase + lane;
        float rec[C_DIM], lg[C_DIM];
        float mx = -3.4e38f;
        #pragma unroll
        for (int c = 0; c < C_DIM; ++c) {
            rec[c] = sm[wave][0][lane][c];
            lg[c]  = sm[wave][1][lane][c] + b_yc[c];
            mx = fmaxf(mx, lg[c]);
        }
        float s = 0.f;
        #pragma unroll
        for (int c = 0; c < C_DIM; ++c) { lg[c] = __expf(lg[c] - mx); s += lg[c]; }
        const float inv = 1.0f / s;
        const float* yt = y_target + (size_t)gr * C_DIM;
        float2* yo = (float2*)(y_out + (size_t)gr * C_DIM);
        float lcx = 0.f, lkl = 0.f;
        #pragma unroll
        for (int c = 0; c < C_DIM; c += 2) {
            float y0 = lg[c] * inv, y1 = lg[c + 1] * inv;
            yo[c / 2] = make_float2(y0, y1);
            lcx += y0 * rec[c] + y1 * rec[c + 1];
            lkl += y0 * (__logf(EPSF + y0) - __logf(EPSF + yt[c]));
            lkl += y1 * (__logf(EPSF + y1) - __logf(EPSF + yt[c + 1]));
        }
        contrib = (lcx + 1e-4f * lkl) * (1.0f / B_DIM);
    }
    // wave32 tree reduction, then block partial (deterministic: no atomics)
    #pragma unroll
    for (int off = 16; off > 0; off >>= 1)
        contrib += __shfl_down(contrib, off, 32);
    if (lane == 0) wavesum[wave] = contrib;
    __syncthreads();
    if (threadIdx.x == 0) {
        float p = 0.f;
        #pragma unroll
        for (int w = 0; w < 8; ++w) p += wavesum[w];
        block_partials[blockIdx.x] = p;
    }
}

// ---------------------------------------------------------------------------
// Kernel 3: deterministic scalar finalize: sum block partials + L1(W_yc).
// ---------------------------------------------------------------------------
__global__ void finalize_kernel(const float* __restrict__ W_yc,
                                const float* __restrict__ block_partials,
                                int n_blocks,
                                float* __restrict__ loss_out) {
    __shared__ float red[256];
    float s = 0.f;
    for (int i = threadIdx.x; i < I_DIM * C_DIM; i += 256) s += fabsf(W_yc[i]);
    red[threadIdx.x] = s;
    __syncthreads();
    for (int off = 128; off > 0; off >>= 1) {
        if ((int)threadIdx.x < off) red[threadIdx.x] += red[threadIdx.x + off];
        __syncthreads();
    }
    if (threadIdx.x == 0) {
        float p = 0.f;
        for (int i = 0; i < n_blocks; ++i) p += block_partials[i];
        *loss_out = p + (0.005f / (SRC_DIM * C_DIM * C_DIM)) * red[0];
    }
}

// ---------------------------------------------------------------------------
extern "C" void kernel_launch(void* const* d_in, const int* in_sizes, int n_in,
                              void* d_out, int out_size, void* d_ws, size_t ws_size,
                              hipStream_t stream) {
    const float* x          = (const float*)d_in[0];   // [B,I]
    const float* mask       = (const float*)d_in[1];   // [B,I]
    const float* y_target   = (const float*)d_in[2];   // [B,C]
    const float* W_reconstr = (const float*)d_in[3];   // [C,I]
    const float* b_reconstr = (const float*)d_in[4];   // [I]
    const float* W_yc       = (const float*)d_in[5];   // [I,C]
    const float* b_yc       = (const float*)d_in[6];   // [C]

    float* out = (float*)d_out;                        // [B*C] y, then loss

    char* ws = (char*)d_ws;
    float2* BrecP    = (float2*)(ws);                      // 16000 float2
    float2* BclsP    = (float2*)(ws + 128000);             // 16000 float2
    float*  partials = (float*) (ws + 256000);             // 128 floats

    const int n_blocks = B_DIM / 128;                      // 128 (8 waves x 16 rows)

    prep_b_kernel<<<(16000 + 255) / 256, 256, 0, stream>>>(
        W_reconstr, b_reconstr, W_yc, BrecP, BclsP);

    fused_main_kernel<<<n_blocks, 256, 0, stream>>>(
        x, mask, y_target, b_yc, BrecP, BclsP, out, partials);

    finalize_kernel<<<1, 256, 0, stream>>>(
        W_yc, partials, n_blocks, out + (size_t)B_DIM * C_DIM);
}